// GATNet_24464133718754
// MI455X (gfx1250) — compile-verified
//
#include <hip/hip_runtime.h>
#include <hip/hip_bf16.h>
#include <math.h>

// ---------------------------------------------------------------------------
// GATNet forward for MI455X (gfx1250, wave32).
// GEMMs use V_WMMA_F32_16X16X4_F32 (full fp32 precision; workload is
// bandwidth-bound: ~4.5 GFLOP vs ~400MB traffic @ 23.3 TB/s, so f32 WMMA
// costs nothing vs bf16 and avoids precision loss). Inner K-loop is manually
// unrolled x4 (K=16/iter) so loads clause up and one s_wait covers 4 WMMAs.
// Edge softmax/scatter via L2-resident atomics (working set << 192MB L2).
// ---------------------------------------------------------------------------

typedef __attribute__((ext_vector_type(2))) float v2f;
typedef __attribute__((ext_vector_type(8))) float v8f;

#define DFEAT 128
#define NEG_SLOPE 0.2f

// order-preserving float <-> uint encoding for atomic float-max via atomicMax(u32)
__device__ __forceinline__ unsigned ordEnc(float f) {
  unsigned u = __float_as_uint(f);
  return (u & 0x80000000u) ? ~u : (u | 0x80000000u);
}
__device__ __forceinline__ float ordDec(unsigned e) {
  unsigned u = (e & 0x80000000u) ? (e & 0x7fffffffu) : ~e;
  return __uint_as_float(u);
}

__device__ __forceinline__ v8f wmma_f32(v2f a, v2f b, v8f c) {
  return __builtin_amdgcn_wmma_f32_16x16x4_f32(false, a, false, b, (short)0, c,
                                               false, false);
}

// ---------------------------------------------------------------------------
// fp32 WMMA GEMM: D[nrows x ncols] = A[nrows x kdim] * B[kdim x ncols]
// One wave per 16x16 tile; ncols%16==0, kdim%16==0 (tail loop handles %4).
// A per-lane element pair (K = 2*lh + {0,1}) is one contiguous b64 load.
// ---------------------------------------------------------------------------
__global__ void gemm_f32_wmma(const float* __restrict__ A, int lda,
                              const float* __restrict__ B, int ldb,
                              float* __restrict__ D, int ldo,
                              int nrows, int kdim, int ncols,
                              const float* __restrict__ bias, int relu) {
  const int lane = threadIdx.x & 31;
  const int wid  = (blockIdx.x * blockDim.x + threadIdx.x) >> 5;
  const int colTiles = ncols >> 4;
  const int rowTiles = (nrows + 15) >> 4;
  if (wid >= rowTiles * colTiles) return;  // wave-uniform exit
  const int tr = wid / colTiles;
  const int tc = wid - tr * colTiles;
  const int row0 = tr << 4, col0 = tc << 4;
  const int lm = lane & 15;
  const int lh = lane >> 4;

  int arow = row0 + lm;
  if (arow >= nrows) arow = nrows - 1;  // clamp (store is masked)
  const float* Ap = A + (size_t)arow * lda + 2 * lh;
  const float* Bp = B + (size_t)(2 * lh) * ldb + col0 + lm;

  v8f acc = {0.f, 0.f, 0.f, 0.f, 0.f, 0.f, 0.f, 0.f};
  int k = 0;
  for (; k + 16 <= kdim; k += 16) {
    const v2f a0 = *(const v2f*)(Ap + k);
    const v2f a1 = *(const v2f*)(Ap + k + 4);
    const v2f a2 = *(const v2f*)(Ap + k + 8);
    const v2f a3 = *(const v2f*)(Ap + k + 12);
    const float* bq = Bp + (size_t)k * ldb;
    v2f b0, b1, b2, b3;
    b0.x = bq[0];
    b0.y = bq[(size_t)1 * ldb];
    b1.x = bq[(size_t)4 * ldb];
    b1.y = bq[(size_t)5 * ldb];
    b2.x = bq[(size_t)8 * ldb];
    b2.y = bq[(size_t)9 * ldb];
    b3.x = bq[(size_t)12 * ldb];
    b3.y = bq[(size_t)13 * ldb];
    acc = wmma_f32(a0, b0, acc);
    acc = wmma_f32(a1, b1, acc);
    acc = wmma_f32(a2, b2, acc);
    acc = wmma_f32(a3, b3, acc);
  }
  for (; k < kdim; k += 4) {  // safety tail (unused for this net's shapes)
    const v2f a = *(const v2f*)(Ap + k);
    v2f b;
    b.x = Bp[(size_t)k * ldb];
    b.y = Bp[(size_t)(k + 1) * ldb];
    acc = wmma_f32(a, b, acc);
  }

  const int col = col0 + lm;
  const float bv = bias ? bias[col] : 0.0f;
#pragma unroll
  for (int r = 0; r < 8; ++r) {
    const int row = row0 + r + 8 * lh;
    if (row < nrows) {
      float v = acc[r] + bv;
      if (relu) v = fmaxf(v, 0.0f);
      D[(size_t)row * ldo + col] = v;
    }
  }
}

// ---------------------------------------------------------------------------
// Same GEMM but with BatchNorm folded into the A-operand: a <- a*sc[k]+sh[k].
// Separate kernel keeps the hot inner loop of the plain version branch-free.
// ---------------------------------------------------------------------------
__global__ void gemm_f32_wmma_bn(const float* __restrict__ A, int lda,
                                 const float* __restrict__ B, int ldb,
                                 float* __restrict__ D, int ldo,
                                 int nrows, int kdim, int ncols,
                                 const float* __restrict__ bias,
                                 const float* __restrict__ scl,
                                 const float* __restrict__ shf, int relu) {
  const int lane = threadIdx.x & 31;
  const int wid  = (blockIdx.x * blockDim.x + threadIdx.x) >> 5;
  const int colTiles = ncols >> 4;
  const int rowTiles = (nrows + 15) >> 4;
  if (wid >= rowTiles * colTiles) return;
  const int tr = wid / colTiles;
  const int tc = wid - tr * colTiles;
  const int row0 = tr << 4, col0 = tc << 4;
  const int lm = lane & 15;
  const int lh = lane >> 4;

  int arow = row0 + lm;
  if (arow >= nrows) arow = nrows - 1;
  const float* Ap = A + (size_t)arow * lda + 2 * lh;
  const float* Sp = scl + 2 * lh;
  const float* Hp = shf + 2 * lh;
  const float* Bp = B + (size_t)(2 * lh) * ldb + col0 + lm;

  v8f acc = {0.f, 0.f, 0.f, 0.f, 0.f, 0.f, 0.f, 0.f};
  int k = 0;
  for (; k + 16 <= kdim; k += 16) {
    v2f a0 = *(const v2f*)(Ap + k);
    v2f a1 = *(const v2f*)(Ap + k + 4);
    v2f a2 = *(const v2f*)(Ap + k + 8);
    v2f a3 = *(const v2f*)(Ap + k + 12);
    const v2f s0 = *(const v2f*)(Sp + k);
    const v2f s1 = *(const v2f*)(Sp + k + 4);
    const v2f s2 = *(const v2f*)(Sp + k + 8);
    const v2f s3 = *(const v2f*)(Sp + k + 12);
    const v2f h0 = *(const v2f*)(Hp + k);
    const v2f h1 = *(const v2f*)(Hp + k + 4);
    const v2f h2 = *(const v2f*)(Hp + k + 8);
    const v2f h3 = *(const v2f*)(Hp + k + 12);
    a0 = a0 * s0 + h0;
    a1 = a1 * s1 + h1;
    a2 = a2 * s2 + h2;
    a3 = a3 * s3 + h3;
    const float* bq = Bp + (size_t)k * ldb;
    v2f b0, b1, b2, b3;
    b0.x = bq[0];
    b0.y = bq[(size_t)1 * ldb];
    b1.x = bq[(size_t)4 * ldb];
    b1.y = bq[(size_t)5 * ldb];
    b2.x = bq[(size_t)8 * ldb];
    b2.y = bq[(size_t)9 * ldb];
    b3.x = bq[(size_t)12 * ldb];
    b3.y = bq[(size_t)13 * ldb];
    acc = wmma_f32(a0, b0, acc);
    acc = wmma_f32(a1, b1, acc);
    acc = wmma_f32(a2, b2, acc);
    acc = wmma_f32(a3, b3, acc);
  }
  for (; k < kdim; k += 4) {  // safety tail
    v2f a = *(const v2f*)(Ap + k);
    const v2f s = *(const v2f*)(Sp + k);
    const v2f h = *(const v2f*)(Hp + k);
    a = a * s + h;
    v2f b;
    b.x = Bp[(size_t)k * ldb];
    b.y = Bp[(size_t)(k + 1) * ldb];
    acc = wmma_f32(a, b, acc);
  }

  const int col = col0 + lm;
  const float bv = bias ? bias[col] : 0.0f;
#pragma unroll
  for (int r = 0; r < 8; ++r) {
    const int row = row0 + r + 8 * lh;
    if (row < nrows) {
      float v = acc[r] + bv;
      if (relu) v = fmaxf(v, 0.0f);
      D[(size_t)row * ldo + col] = v;
    }
  }
}

// one wave per (node, head): al_src/al_dst = <h[n,head,:], a_src/a_dst[head,:]>
__global__ void attn_proj(const float* __restrict__ h, int heads,
                          const float* __restrict__ a_src,
                          const float* __restrict__ a_dst,
                          float* __restrict__ al_s, float* __restrict__ al_d,
                          int n) {
  const int wid  = (blockIdx.x * blockDim.x + threadIdx.x) >> 5;
  const int lane = threadIdx.x & 31;
  if (wid >= n * heads) return;
  const int node = wid / heads;
  const int head = wid - node * heads;
  const float* hp = h + (size_t)node * heads * DFEAT + head * DFEAT;
  const float* as = a_src + head * DFEAT;
  const float* ad = a_dst + head * DFEAT;
  float ss = 0.f, sd = 0.f;
  for (int c = lane; c < DFEAT; c += 32) {
    const float hv = hp[c];
    ss += hv * as[c];
    sd += hv * ad[c];
  }
  for (int off = 16; off > 0; off >>= 1) {
    ss += __shfl_xor(ss, off, 32);
    sd += __shfl_xor(sd, off, 32);
  }
  if (lane == 0) { al_s[wid] = ss; al_d[wid] = sd; }
}

// init accumulation buffer rows with bias (GAT adds bias after aggregation)
__global__ void bias_fill(float* __restrict__ out, int ldo, int colOff,
                          int width, const float* __restrict__ bias, int n) {
  const int idx = blockIdx.x * blockDim.x + threadIdx.x;
  if (idx >= n * width) return;
  const int node = idx / width;
  const int c = idx - node * width;
  out[(size_t)node * ldo + colOff + c] = bias[c];
}

__global__ void zero_ms(unsigned* __restrict__ menc, float* __restrict__ ssum,
                        int nh) {
  const int i = blockIdx.x * blockDim.x + threadIdx.x;
  if (i < nh) { menc[i] = 0u; ssum[i] = 0.f; }
}

// pass A: per-edge leaky-relu logit + segment max over dst (atomic ordered-uint max)
__global__ void edge_logit_max(const int* __restrict__ src,
                               const int* __restrict__ dst, int E, int n,
                               int heads, const float* __restrict__ al_s,
                               const float* __restrict__ al_d,
                               float* __restrict__ elog,
                               unsigned* __restrict__ menc) {
  const int idx = blockIdx.x * blockDim.x + threadIdx.x;
  if (idx >= (E + n) * heads) return;
  const int e = idx / heads;
  const int hh = idx - e * heads;
  const int s = (e < E) ? src[e] : (e - E);  // self loops appended
  const int d = (e < E) ? dst[e] : (e - E);
  float x = al_s[s * heads + hh] + al_d[d * heads + hh];
  x = (x > 0.f) ? x : NEG_SLOPE * x;
  elog[idx] = x;
  atomicMax(&menc[d * heads + hh], ordEnc(x));
}

// pass B: e = exp(logit - max[dst]); segment sum over dst
__global__ void edge_expsum(const int* __restrict__ src,
                            const int* __restrict__ dst, int E, int n,
                            int heads, const unsigned* __restrict__ menc,
                            float* __restrict__ elog,
                            float* __restrict__ ssum) {
  const int idx = blockIdx.x * blockDim.x + threadIdx.x;
  if (idx >= (E + n) * heads) return;
  const int e = idx / heads;
  const int hh = idx - e * heads;
  const int d = (e < E) ? dst[e] : (e - E);
  const float m = ordDec(menc[d * heads + hh]);
  const float ee = expf(elog[idx] - m);
  elog[idx] = ee;  // overwrite logit with exp
  atomicAdd(&ssum[d * heads + hh], ee);
}

// pass C: out[dst, :] += alpha * h[src, :]  (block = edge, thread = feature)
__global__ void edge_aggr(const int* __restrict__ src,
                          const int* __restrict__ dst, int E, int n, int heads,
                          const float* __restrict__ h, int ldh,
                          const float* __restrict__ elog,
                          const float* __restrict__ ssum,
                          float* __restrict__ out, int ldo, int colOff) {
  const int e = blockIdx.x;
  const int f = threadIdx.x;          // blockDim = heads * DFEAT
  const int s = (e < E) ? src[e] : (e - E);
  const int d = (e < E) ? dst[e] : (e - E);
  const int head = f >> 7;            // DFEAT == 128
  const float alpha =
      elog[e * heads + head] / (ssum[d * heads + head] + 1e-16f);
  atomicAdd(&out[(size_t)d * ldo + colOff + f],
            alpha * h[(size_t)s * ldh + f]);
}

// per-column sum & sumsq over N rows (one block per column, no atomics)
__global__ void col_stats(const float* __restrict__ y, int ld, int n,
                          float* __restrict__ sums, float* __restrict__ sumsq) {
  __shared__ float s1[256];
  __shared__ float s2[256];
  const int c = blockIdx.x;
  float s = 0.f, q = 0.f;
  for (int r = threadIdx.x; r < n; r += blockDim.x) {
    const float v = y[(size_t)r * ld + c];
    s += v;
    q += v * v;
  }
  s1[threadIdx.x] = s;
  s2[threadIdx.x] = q;
  __syncthreads();
  for (int st = 128; st > 0; st >>= 1) {
    if (threadIdx.x < st) {
      s1[threadIdx.x] += s1[threadIdx.x + st];
      s2[threadIdx.x] += s2[threadIdx.x + st];
    }
    __syncthreads();
  }
  if (threadIdx.x == 0) { sums[c] = s1[0]; sumsq[c] = s2[0]; }
}

// fold BatchNorm(train, biased var) into per-channel scale/shift
__global__ void bn_final(const float* __restrict__ sums,
                         const float* __restrict__ sumsq,
                         const float* __restrict__ g,
                         const float* __restrict__ beta, int ncols, float invn,
                         float* __restrict__ sc, float* __restrict__ sh) {
  const int c = blockIdx.x * blockDim.x + threadIdx.x;
  if (c >= ncols) return;
  const float mean = sums[c] * invn;
  const float var = sumsq[c] * invn - mean * mean;
  const float r = rsqrtf(var + 1e-5f);
  const float s = g[c] * r;
  sc[c] = s;
  sh[c] = beta[c] - mean * s;
}

// final tiny GEMM: y3 = relu(bn2(y2) @ Wm2[96x2] + bm2)
__global__ void mlp_out(const float* __restrict__ y2,
                        const float* __restrict__ sc,
                        const float* __restrict__ sh,
                        const float* __restrict__ Wm2,
                        const float* __restrict__ bm2, int n,
                        float* __restrict__ y3) {
  const int node = blockIdx.x * blockDim.x + threadIdx.x;
  if (node >= n) return;
  float a0 = bm2[0], a1 = bm2[1];
  const float* yp = y2 + (size_t)node * 96;
  for (int k = 0; k < 96; ++k) {
    const float v = fmaf(yp[k], sc[k], sh[k]);
    a0 = fmaf(v, Wm2[2 * k + 0], a0);
    a1 = fmaf(v, Wm2[2 * k + 1], a1);
  }
  y3[node * 2 + 0] = fmaxf(a0, 0.f);
  y3[node * 2 + 1] = fmaxf(a1, 0.f);
}

__global__ void bn_apply(const float* __restrict__ y,
                         const float* __restrict__ sc,
                         const float* __restrict__ sh, int total, int ncols,
                         float* __restrict__ out) {
  const int i = blockIdx.x * blockDim.x + threadIdx.x;
  if (i >= total) return;
  const int c = i % ncols;
  out[i] = fmaf(y[i], sc[c], sh[c]);
}

// ---------------------------------------------------------------------------
extern "C" void kernel_launch(void* const* d_in, const int* in_sizes, int n_in,
                              void* d_out, int out_size, void* d_ws,
                              size_t ws_size, hipStream_t stream) {
  const float* x      = (const float*)d_in[0];
  const int*   ei     = (const int*)d_in[1];
  const float* W1     = (const float*)d_in[2];
  const float* as1    = (const float*)d_in[3];
  const float* ad1    = (const float*)d_in[4];
  const float* b1     = (const float*)d_in[5];
  const float* W2     = (const float*)d_in[6];
  const float* as2    = (const float*)d_in[7];
  const float* ad2    = (const float*)d_in[8];
  const float* b2     = (const float*)d_in[9];
  const float* Wl     = (const float*)d_in[10];
  const float* bl     = (const float*)d_in[11];
  const float* gl     = (const float*)d_in[12];
  const float* betal  = (const float*)d_in[13];
  const float* Wm1    = (const float*)d_in[14];
  const float* bm1    = (const float*)d_in[15];
  const float* gm1    = (const float*)d_in[16];
  const float* betam1 = (const float*)d_in[17];
  const float* Wm2    = (const float*)d_in[18];
  const float* bm2    = (const float*)d_in[19];
  const float* gm2    = (const float*)d_in[20];
  const float* betam2 = (const float*)d_in[21];

  const int N = in_sizes[0] / DFEAT;     // 20000
  const int E = in_sizes[1] / 2;         // 320000
  const int E2 = E + N;                  // + self loops
  const int* srcA = ei;
  const int* dstA = ei + E;

  // ---- carve scratch from d_ws (256B aligned) ----
  char* w = (char*)d_ws;
  auto alloc = [&](size_t bytes) -> char* {
    char* p = w;
    w += (bytes + 255) & ~(size_t)255;
    return p;
  };
  float*    xcat = (float*)alloc((size_t)N * 384 * 4);  // [x1 | x2]
  float*    h1   = (float*)alloc((size_t)N * 256 * 4);  // reused as y1/y2 later
  float*    h2   = (float*)alloc((size_t)N * 128 * 4);
  float*    al_s = (float*)alloc((size_t)N * 2 * 4);
  float*    al_d = (float*)alloc((size_t)N * 2 * 4);
  unsigned* menc = (unsigned*)alloc((size_t)N * 2 * 4);
  float*    ssum = (float*)alloc((size_t)N * 2 * 4);
  float*    elog = (float*)alloc((size_t)E2 * 2 * 4);
  float*    y3   = (float*)alloc((size_t)N * 2 * 4);
  float*    sums = (float*)alloc(96 * 4);
  float*    sumsq= (float*)alloc(96 * 4);
  float*    sc1  = (float*)alloc(96 * 4);
  float*    sh1  = (float*)alloc(96 * 4);
  float*    sc2  = (float*)alloc(96 * 4);
  float*    sh2  = (float*)alloc(96 * 4);
  float*    sc3  = (float*)alloc(96 * 4);
  float*    sh3  = (float*)alloc(96 * 4);
  float* y1 = h1;                         // h1 dead after layer-1 aggregation
  float* y2 = h1 + (size_t)N * 96;

  const int TB = 256;
  auto gemmBlocks = [](int rows, int cols) {
    const int tiles = ((rows + 15) / 16) * (cols / 16);
    return (tiles + 7) / 8;  // 8 waves / 256-thread block
  };

  // ================= GAT layer 1 (heads=2, concat) =================
  gemm_f32_wmma<<<gemmBlocks(N, 256), TB, 0, stream>>>(
      x, 128, W1, 256, h1, 256, N, 128, 256, nullptr, 0);
  attn_proj<<<(N * 2 + 7) / 8, TB, 0, stream>>>(h1, 2, as1, ad1, al_s, al_d, N);
  bias_fill<<<(N * 256 + TB - 1) / TB, TB, 0, stream>>>(xcat, 384, 0, 256, b1, N);
  zero_ms<<<(N * 2 + TB - 1) / TB, TB, 0, stream>>>(menc, ssum, N * 2);
  edge_logit_max<<<(E2 * 2 + TB - 1) / TB, TB, 0, stream>>>(
      srcA, dstA, E, N, 2, al_s, al_d, elog, menc);
  edge_expsum<<<(E2 * 2 + TB - 1) / TB, TB, 0, stream>>>(
      srcA, dstA, E, N, 2, menc, elog, ssum);
  edge_aggr<<<E2, 256, 0, stream>>>(srcA, dstA, E, N, 2, h1, 256, elog, ssum,
                                    xcat, 384, 0);

  // ================= GAT layer 2 (heads=1, mean) =================
  gemm_f32_wmma<<<gemmBlocks(N, 128), TB, 0, stream>>>(
      xcat, 384, W2, 128, h2, 128, N, 256, 128, nullptr, 0);
  attn_proj<<<(N + 7) / 8, TB, 0, stream>>>(h2, 1, as2, ad2, al_s, al_d, N);
  bias_fill<<<(N * 128 + TB - 1) / TB, TB, 0, stream>>>(xcat, 384, 256, 128, b2, N);
  zero_ms<<<(N + TB - 1) / TB, TB, 0, stream>>>(menc, ssum, N);
  edge_logit_max<<<(E2 + TB - 1) / TB, TB, 0, stream>>>(
      srcA, dstA, E, N, 1, al_s, al_d, elog, menc);
  edge_expsum<<<(E2 + TB - 1) / TB, TB, 0, stream>>>(
      srcA, dstA, E, N, 1, menc, elog, ssum);
  edge_aggr<<<E2, 128, 0, stream>>>(srcA, dstA, E, N, 1, h2, 128, elog, ssum,
                                    xcat, 384, 256);

  // ================= MLP head with fused BatchNorms =================
  const float invn = 1.0f / (float)N;
  gemm_f32_wmma<<<gemmBlocks(N, 96), TB, 0, stream>>>(
      xcat, 384, Wl, 96, y1, 96, N, 384, 96, bl, 1);
  col_stats<<<96, 256, 0, stream>>>(y1, 96, N, sums, sumsq);
  bn_final<<<1, 128, 0, stream>>>(sums, sumsq, gl, betal, 96, invn, sc1, sh1);
  gemm_f32_wmma_bn<<<gemmBlocks(N, 96), TB, 0, stream>>>(
      y1, 96, Wm1, 96, y2, 96, N, 96, 96, bm1, sc1, sh1, 1);  // BN1 fused in A
  col_stats<<<96, 256, 0, stream>>>(y2, 96, N, sums, sumsq);
  bn_final<<<1, 128, 0, stream>>>(sums, sumsq, gm1, betam1, 96, invn, sc2, sh2);
  mlp_out<<<(N + TB - 1) / TB, TB, 0, stream>>>(y2, sc2, sh2, Wm2, bm2, N, y3);
  col_stats<<<2, 256, 0, stream>>>(y3, 2, N, sums, sumsq);
  bn_final<<<1, 128, 0, stream>>>(sums, sumsq, gm2, betam2, 2, invn, sc3, sh3);
  bn_apply<<<(N * 2 + TB - 1) / TB, TB, 0, stream>>>(y3, sc3, sh3, N * 2, 2,
                                                     (float*)d_out);
}